// SlotAttentionModule_89197880803348
// MI455X (gfx1250) — compile-verified
//
#include <hip/hip_runtime.h>
#include <math.h>

typedef float v2f __attribute__((ext_vector_type(2)));
typedef float v4f __attribute__((ext_vector_type(4)));
typedef float v8f __attribute__((ext_vector_type(8)));

#define BATCH 16
#define NTOK  16384
#define DINC  128
#define KSLOT 8
#define SDIM  64
#define HDIM  128
#define NITER 3

// ---------------- workspace layout (in floats) ----------------
// kbuf : BATCH*NTOK*SDIM            = 16777216
// vbuf : BATCH*NTOK*SDIM            = 16777216
// qbuf : BATCH*KSLOT*SDIM (+512 pad, A-frag rows 8..15 over-read)
// upd  : BATCH*KSLOT*SDIM
// sums : BATCH*KSLOT
#define WS_KBUF 0L
#define WS_VBUF (WS_KBUF + (long)BATCH*NTOK*SDIM)
#define WS_QBUF (WS_VBUF + (long)BATCH*NTOK*SDIM)
#define WS_UPD  (WS_QBUF + (long)BATCH*KSLOT*SDIM + 512)
#define WS_SUMS (WS_UPD  + (long)BATCH*KSLOT*SDIM)

// ---------------- slots init: mu + exp(log_sigma)*noise ----------------
__global__ __launch_bounds__(256) void init_slots_kernel(
    const float* __restrict__ noise, const float* __restrict__ mu,
    const float* __restrict__ lsig, float* __restrict__ slots)
{
  int e = blockIdx.x * 256 + threadIdx.x;           // 0..8191
  int s = e & (SDIM - 1);
  slots[e] = mu[s] + __expf(lsig[s]) * noise[e];
}

// ---------------- fused LN(inputs) + k/v projection -------------------
// grid: B*N/16 blocks of 256 (8 waves). waves 0-3: k cols, waves 4-7: v cols.
__global__ __launch_bounds__(256) void kv_proj_kernel(
    const float* __restrict__ inputs, const float* __restrict__ lng,
    const float* __restrict__ lnb, const float* __restrict__ Wk,
    const float* __restrict__ Wv, float* __restrict__ kbuf,
    float* __restrict__ vbuf)
{
  __shared__ float xs[16][132];                     // stride 132 % 64 banks = 4
  const int tid = threadIdx.x;
  const long rowBase = (long)blockIdx.x * 16;

  // ---- LayerNorm of 16 rows x 128 cols: 16 threads per row, 8 elems each ----
  {
    int row = tid >> 4, seg = tid & 15;
    const v4f* src = (const v4f*)(inputs + (rowBase + row) * DINC + seg * 8);
    v4f x0 = __builtin_nontemporal_load(src);       // inputs read exactly once:
    v4f x1 = __builtin_nontemporal_load(src + 1);   // NT keeps k/v resident in L2
    float x[8] = {x0.x, x0.y, x0.z, x0.w, x1.x, x1.y, x1.z, x1.w};
    float s = 0.f, s2 = 0.f;
#pragma unroll
    for (int i = 0; i < 8; ++i) { s += x[i]; s2 += x[i] * x[i]; }
#pragma unroll
    for (int m = 8; m >= 1; m >>= 1) {
      s  += __shfl_xor(s,  m, 16);
      s2 += __shfl_xor(s2, m, 16);
    }
    float mean = s * (1.f / 128.f);
    float var  = s2 * (1.f / 128.f) - mean * mean;
    float inv  = rsqrtf(var + 1e-5f);
#pragma unroll
    for (int i = 0; i < 8; ++i) {
      int col = seg * 8 + i;
      xs[row][col] = (x[i] - mean) * inv * lng[col] + lnb[col];
    }
  }
  __syncthreads();

  // ---- WMMA f32 16x16x4: D(16x16) += A(16x4) * B(4x16), 32 K-steps ----
  const int wave  = tid >> 5, lane = tid & 31;
  const int n     = lane & 15;          // A row m / B col n
  const int khalf = lane >> 4;          // 0: K={0,1}, 1: K={2,3}
  const float* W  = (wave < 4) ? Wk : Wv;
  float* out      = (wave < 4) ? kbuf : vbuf;
  const int colBase = (wave & 3) * 16;
  const v2f* wrow = (const v2f*)(W + (long)(colBase + n) * DINC + khalf * 2);

  v8f c = {0.f, 0.f, 0.f, 0.f, 0.f, 0.f, 0.f, 0.f};
#pragma unroll 4
  for (int kk = 0; kk < 32; ++kk) {
    v2f a, b;
    a.x = xs[n][kk * 4 + khalf * 2 + 0];
    a.y = xs[n][kk * 4 + khalf * 2 + 1];
    b   = wrow[kk * 2];
    c = __builtin_amdgcn_wmma_f32_16x16x4_f32(false, a, false, b,
                                              (short)0, c, false, false);
  }
#pragma unroll
  for (int i = 0; i < 8; ++i) {
    long r = rowBase + i + 8 * khalf;               // D: m = vgpr + 8*(lane/16)
    out[r * SDIM + colBase + n] = c[i];
  }
}

// ---------------- per-iter: LN(slots) @ Wq^T, zero accumulators -------
// grid: 128 blocks (one per (b,slot) row) of 64 threads
__global__ __launch_bounds__(64) void q_kernel(
    const float* __restrict__ slots, const float* __restrict__ lng,
    const float* __restrict__ lnb, const float* __restrict__ Wq,
    float* __restrict__ qbuf, float* __restrict__ upd,
    float* __restrict__ sums)
{
  __shared__ float sv[SDIM];
  __shared__ float red[SDIM];
  __shared__ float stats[2];
  int r = blockIdx.x, t = threadIdx.x;
  float val = slots[r * SDIM + t];
  red[t] = val;
  __syncthreads();
  if (t == 0) { float s = 0.f; for (int i = 0; i < SDIM; ++i) s += red[i]; stats[0] = s * (1.f / 64.f); }
  __syncthreads();
  float d = val - stats[0];
  red[t] = d * d;
  __syncthreads();
  if (t == 0) { float s = 0.f; for (int i = 0; i < SDIM; ++i) s += red[i]; stats[1] = rsqrtf(s * (1.f / 64.f) + 1e-5f); }
  __syncthreads();
  sv[t] = d * stats[1] * lng[t] + lnb[t];
  upd[r * SDIM + t] = 0.f;
  if (t == 0) sums[r] = 0.f;
  __syncthreads();
  float acc = 0.f;
  const float* wr = Wq + t * SDIM;
#pragma unroll 8
  for (int j = 0; j < SDIM; ++j) acc += wr[j] * sv[j];
  qbuf[r * SDIM + t] = acc;
}

// -------- per-iter FUSED: logits -> softmax -> token sums + attn^T @ v --------
// grid: 16 batches * 32 blocks of 256 (8 waves). Each wave: 64 tokens (4 tiles).
// Denominator (token renorm) applied later in slot_kernel (linear in attn).
__global__ __launch_bounds__(256) void attn_upd_kernel(
    const float* __restrict__ kbuf, const float* __restrict__ vbuf,
    const float* __restrict__ qbuf, float* __restrict__ upd,
    float* __restrict__ sums)
{
  __shared__ float atile[8][16][17];                // [wave][token][slot(pad 17)]
  __shared__ float psum[8][KSLOT];
  const int tid = threadIdx.x, wave = tid >> 5, lane = tid & 31;
  const int b     = blockIdx.x >> 5;                // 32 blocks per batch
  const int split = (blockIdx.x & 31) * 8 + wave;   // 0..255, 64 tokens each
  const long tokBase0 = (long)b * NTOK + (long)split * 64;
  const int m = lane & 15, khalf = lane >> 4;

  // zero pad slots 8..15 of this wave's staging tile (A rows 8..15 -> D rows
  // 8..15 become exact zeros; those lanes never store anyway)
  if (lane < 16) {
#pragma unroll
    for (int i = 0; i < 8; ++i) atile[wave][lane][8 + i] = 0.f;
  }

  // preload q fragments (constant per batch): A[m=slot][k], rows 8..15 read pad
  v2f qfrag[16];
  const v2f* qp = (const v2f*)(qbuf + (long)b * KSLOT * SDIM + m * SDIM + khalf * 2);
#pragma unroll
  for (int kk = 0; kk < 16; ++kk) qfrag[kk] = qp[kk * 2];

  v8f cu[4];                                        // update accumulators, 4 d-tiles
#pragma unroll
  for (int ct = 0; ct < 4; ++ct) cu[ct] = (v8f){0.f,0.f,0.f,0.f,0.f,0.f,0.f,0.f};
  float sacc[KSLOT] = {0.f,0.f,0.f,0.f,0.f,0.f,0.f,0.f};
  const float scale = 0.125f;                       // 64^-0.5

  for (int t = 0; t < 4; ++t) {
    const long tokBase = tokBase0 + t * 16;

    // ---- phase 1: logits tile via WMMA (A=q, B=k^T tile) ----
    const v2f* kp = (const v2f*)(kbuf + (tokBase + m) * SDIM + khalf * 2);
    v8f cl = {0.f, 0.f, 0.f, 0.f, 0.f, 0.f, 0.f, 0.f};
#pragma unroll
    for (int kk = 0; kk < 16; ++kk) {
      v2f bf = kp[kk * 2];
      cl = __builtin_amdgcn_wmma_f32_16x16x4_f32(false, qfrag[kk], false, bf,
                                                 (short)0, cl, false, false);
    }
    // lanes 0..15: cl[i] = logit[slot i][token = tokBase+lane]
    float mx = -1e30f, l[KSLOT];
#pragma unroll
    for (int i = 0; i < KSLOT; ++i) { l[i] = cl[i] * scale; mx = fmaxf(mx, l[i]); }
    float sum = 0.f, at[KSLOT];
#pragma unroll
    for (int i = 0; i < KSLOT; ++i) { at[i] = __expf(l[i] - mx); sum += at[i]; }
    float rinv = 1.f / sum;
#pragma unroll
    for (int i = 0; i < KSLOT; ++i) { at[i] *= rinv; sacc[i] += at[i]; }

    if (lane < 16) {
#pragma unroll
      for (int i = 0; i < KSLOT; ++i) atile[wave][lane][i] = at[i];
    }
    __syncthreads();

    // ---- phase 2: cu[ct] += attn^T(8x16) @ v(16x16-col-tile) ----
#pragma unroll
    for (int kk2 = 0; kk2 < 4; ++kk2) {
      const int t0 = kk2 * 4 + khalf * 2;
      v2f a2;
      a2.x = atile[wave][t0 + 0][m];
      a2.y = atile[wave][t0 + 1][m];
      const float* v0 = vbuf + (tokBase + t0) * SDIM + m;
#pragma unroll
      for (int ct = 0; ct < 4; ++ct) {
        v2f bf;
        bf.x = v0[ct * 16];
        bf.y = v0[ct * 16 + SDIM];
        cu[ct] = __builtin_amdgcn_wmma_f32_16x16x4_f32(false, a2, false, bf,
                                                       (short)0, cu[ct], false, false);
      }
    }
    __syncthreads();                                // before next tile's atile write
  }

  // ---- token-sum reduction: lanes 0..15 hold valid per-token attn ----
#pragma unroll
  for (int i = 0; i < KSLOT; ++i) {
#pragma unroll
    for (int s = 8; s >= 1; s >>= 1) sacc[i] += __shfl_xor(sacc[i], s, 16);
  }
  if (lane == 0) {
#pragma unroll
    for (int i = 0; i < KSLOT; ++i) psum[wave][i] = sacc[i];
  }
  __syncthreads();
  if (tid < KSLOT) {
    float tot = 0.f;
#pragma unroll
    for (int w = 0; w < 8; ++w) tot += psum[w][tid];
    atomicAdd(&sums[b * KSLOT + tid], tot);
  }

  // ---- raw update accumulation: D[m=slot i][n=d col] for lanes < 16 ----
  if (lane < 16) {
#pragma unroll
    for (int ct = 0; ct < 4; ++ct) {
#pragma unroll
      for (int i = 0; i < KSLOT; ++i)
        atomicAdd(&upd[(long)b * KSLOT * SDIM + i * SDIM + ct * 16 + lane], cu[ct][i]);
    }
  }
}

// ---------------- per-iter: denom + GRU + LN + MLP slot update --------
// grid: 128 blocks (one per (b,slot)) of 64 threads
__global__ __launch_bounds__(64) void slot_kernel(
    float* __restrict__ slots, const float* __restrict__ upd,
    const float* __restrict__ sums,
    const float* __restrict__ Wih, const float* __restrict__ Whh,
    const float* __restrict__ bih, const float* __restrict__ bhh,
    const float* __restrict__ lng, const float* __restrict__ lnb,
    const float* __restrict__ W1, const float* __restrict__ b1,
    const float* __restrict__ W2, const float* __restrict__ b2)
{
  __shared__ float u[SDIM], h[SDIM], mvec[SDIM], hid[HDIM];
  __shared__ float red[SDIM];
  __shared__ float stats[2];
  const int r = blockIdx.x, t = threadIdx.x;
  const float denom = sums[r] + 1e-8f;
  u[t] = upd[r * SDIM + t] / denom;
  const float hv = slots[r * SDIM + t];
  h[t] = hv;
  __syncthreads();

  float gi_r = bih[t], gi_z = bih[SDIM + t], gi_n = bih[2 * SDIM + t];
  float gh_r = bhh[t], gh_z = bhh[SDIM + t], gh_n = bhh[2 * SDIM + t];
  const float* wr = Wih + (long)t * SDIM;
  const float* wz = Wih + (long)(SDIM + t) * SDIM;
  const float* wn = Wih + (long)(2 * SDIM + t) * SDIM;
  const float* vr = Whh + (long)t * SDIM;
  const float* vz = Whh + (long)(SDIM + t) * SDIM;
  const float* vn = Whh + (long)(2 * SDIM + t) * SDIM;
#pragma unroll 4
  for (int j = 0; j < SDIM; ++j) {
    gi_r += wr[j] * u[j];  gi_z += wz[j] * u[j];  gi_n += wn[j] * u[j];
    gh_r += vr[j] * h[j];  gh_z += vz[j] * h[j];  gh_n += vn[j] * h[j];
  }
  const float rg = 1.f / (1.f + __expf(-(gi_r + gh_r)));
  const float zg = 1.f / (1.f + __expf(-(gi_z + gh_z)));
  const float ng = tanhf(gi_n + rg * gh_n);
  const float nsv = (1.f - zg) * ng + zg * hv;

  red[t] = nsv;
  __syncthreads();
  if (t == 0) { float s = 0.f; for (int i = 0; i < SDIM; ++i) s += red[i]; stats[0] = s * (1.f / 64.f); }
  __syncthreads();
  const float dd = nsv - stats[0];
  red[t] = dd * dd;
  __syncthreads();
  if (t == 0) { float s = 0.f; for (int i = 0; i < SDIM; ++i) s += red[i]; stats[1] = rsqrtf(s * (1.f / 64.f) + 1e-5f); }
  __syncthreads();
  mvec[t] = dd * stats[1] * lng[t] + lnb[t];
  __syncthreads();

  float a0 = b1[t], a1 = b1[SDIM + t];
  const float* w1a = W1 + (long)t * SDIM;
  const float* w1b = W1 + (long)(SDIM + t) * SDIM;
#pragma unroll 4
  for (int j = 0; j < SDIM; ++j) { a0 += w1a[j] * mvec[j]; a1 += w1b[j] * mvec[j]; }
  hid[t] = fmaxf(a0, 0.f);
  hid[SDIM + t] = fmaxf(a1, 0.f);
  __syncthreads();

  float o = b2[t];
  const float* w2 = W2 + (long)t * HDIM;
#pragma unroll 4
  for (int j = 0; j < HDIM; ++j) o += w2[j] * hid[j];
  slots[r * SDIM + t] = nsv + o;
}

// ----------------------------------------------------------------------
extern "C" void kernel_launch(void* const* d_in, const int* in_sizes, int n_in,
                              void* d_out, int out_size, void* d_ws, size_t ws_size,
                              hipStream_t stream) {
  const float* inputs     = (const float*)d_in[0];
  const float* slot_noise = (const float*)d_in[1];
  const float* slot_mu    = (const float*)d_in[2];
  const float* slot_lsig  = (const float*)d_in[3];
  const float* ln_in_g    = (const float*)d_in[4];
  const float* ln_in_b    = (const float*)d_in[5];
  const float* ln_s_g     = (const float*)d_in[6];
  const float* ln_s_b     = (const float*)d_in[7];
  const float* ln_m_g     = (const float*)d_in[8];
  const float* ln_m_b     = (const float*)d_in[9];
  const float* Wk         = (const float*)d_in[10];
  const float* Wv         = (const float*)d_in[11];
  const float* Wq         = (const float*)d_in[12];
  const float* gru_Wih    = (const float*)d_in[13];
  const float* gru_Whh    = (const float*)d_in[14];
  const float* gru_bih    = (const float*)d_in[15];
  const float* gru_bhh    = (const float*)d_in[16];
  const float* mlp_W1     = (const float*)d_in[17];
  const float* mlp_b1     = (const float*)d_in[18];
  const float* mlp_W2     = (const float*)d_in[19];
  const float* mlp_b2     = (const float*)d_in[20];

  float* slots = (float*)d_out;                     // (B,K,S) lives in d_out
  float* ws    = (float*)d_ws;
  float* kbuf  = ws + WS_KBUF;
  float* vbuf  = ws + WS_VBUF;
  float* qbuf  = ws + WS_QBUF;
  float* upd   = ws + WS_UPD;
  float* sums  = ws + WS_SUMS;

  init_slots_kernel<<<(BATCH * KSLOT * SDIM) / 256, 256, 0, stream>>>(
      slot_noise, slot_mu, slot_lsig, slots);

  kv_proj_kernel<<<(BATCH * NTOK) / 16, 256, 0, stream>>>(
      inputs, ln_in_g, ln_in_b, Wk, Wv, kbuf, vbuf);

  for (int it = 0; it < NITER; ++it) {
    q_kernel<<<BATCH * KSLOT, 64, 0, stream>>>(
        slots, ln_s_g, ln_s_b, Wq, qbuf, upd, sums);
    attn_upd_kernel<<<BATCH * 32, 256, 0, stream>>>(
        kbuf, vbuf, qbuf, upd, sums);
    slot_kernel<<<BATCH * KSLOT, 64, 0, stream>>>(
        slots, upd, sums, gru_Wih, gru_Whh, gru_bih, gru_bhh,
        ln_m_g, ln_m_b, mlp_W1, mlp_b1, mlp_W2, mlp_b2);
  }
}